// Mat2Twist_1949915152783
// MI455X (gfx1250) — compile-verified
//
#include <hip/hip_runtime.h>
#include <cstdint>

#define TILE 256          // matrices per block
#define TPB  256          // threads per block (8 wave32 waves)
// Per wave: 32 matrices = 288 dwords = 1152 bytes of contiguous input.

__device__ __forceinline__ void twist_math_store(
    float m00, float m01, float m02,
    float m10, float m11, float m12,
    float m20, float m21, float m22,
    float* __restrict__ out, size_t gid)
{
    // theta = acos((tr-1)/2);  2*sin(acos(c)) == 2*sqrt(1-c^2)
    float c = 0.5f * (m00 + m11 + m22 - 1.0f);
    c = fminf(1.0f, fmaxf(-1.0f, c));
    float theta   = acosf(c);
    float s_theta = 2.0f * sqrtf(fmaxf(1.0f - c * c, 0.0f));
    float f       = theta / s_theta;   // theta in (0.1, pi-0.1): never 0/0

    float* o = out + gid * 3;
    o[0] = f * (m21 - m12);
    o[1] = f * (m02 - m20);
    o[2] = f * (m10 - m01);
}

// Straight-line hot kernel: every block handles a full tile of 256 matrices.
// Each wave stages ONLY its own 32 matrices via async global->LDS DMA, so the
// producer and consumer are the same wave: s_wait_asynccnt 0 is sufficient,
// no workgroup barrier at all.
__global__ __launch_bounds__(TPB)
void mat2twist_main(const float* __restrict__ in, float* __restrict__ out)
{
    __shared__ float smem[TILE * 9];   // 9216 B; 1152 B slice per wave

    const int tid  = threadIdx.x;
    const int wid  = tid >> 5;         // wave id within block (0..7)
    const int lane = tid & 31;
    const int blk  = blockIdx.x;

    const float* gbase = in + (size_t)blk * (TILE * 9);   // uniform -> SGPR pair

    // Low 32 bits of a generic LDS pointer == byte offset within the LDS
    // allocation; the async instruction adds the wave's LDS_BASE itself.
    const uint32_t lds_base = (uint32_t)(uintptr_t)(&smem[0]);
    const uint32_t wbyte    = (uint32_t)(wid * 288 + lane) * 4u; // wave slice + lane dword
    const uint32_t lds_wr   = lds_base + wbyte;
    const uint32_t voff     = wbyte;

    // GVS mode: global = SADDR + VADDR + offset ; LDS = VDST + offset.
    // offset:i*128 advances both sides together:
    //   lds[wid*288 + i*32 + lane] = g[wid*288 + i*32 + lane],  i = 0..8
    // Each instruction moves a contiguous 128 B per wave. The async op does
    // not write any VGPR (VDST only supplies the LDS address), so reusing the
    // input registers across all 9 instructions is safe.
    asm volatile(
        "global_load_async_to_lds_b32 %0, %1, %2 offset:0\n\t"
        "global_load_async_to_lds_b32 %0, %1, %2 offset:128\n\t"
        "global_load_async_to_lds_b32 %0, %1, %2 offset:256\n\t"
        "global_load_async_to_lds_b32 %0, %1, %2 offset:384\n\t"
        "global_load_async_to_lds_b32 %0, %1, %2 offset:512\n\t"
        "global_load_async_to_lds_b32 %0, %1, %2 offset:640\n\t"
        "global_load_async_to_lds_b32 %0, %1, %2 offset:768\n\t"
        "global_load_async_to_lds_b32 %0, %1, %2 offset:896\n\t"
        "global_load_async_to_lds_b32 %0, %1, %2 offset:1024\n\t"
        "s_wait_asynccnt 0"
        :
        : "v"(lds_wr), "v"(voff), "s"(gbase)
        : "memory");

    // Consume the wave's own slice: lane reads its 9 contiguous floats.
    // Stride 9 dwords across 64 banks (gcd(9,64)=1) -> conflict-free.
    //
    // Outputs are EARLYCLOBBER ("=&v"): the block contains multiple
    // instructions, so results must not share registers with the address
    // input %9 (a plain "=v" let the allocator emit "ds_load_b32 v1, v1",
    // clobbering the address mid-sequence).
    float m00, m01, m02, m10, m11, m12, m20, m21, m22;
    const uint32_t lds_rd = lds_base + (uint32_t)(wid * 1152 + lane * 36);
    asm volatile(
        "ds_load_b32 %0, %9 offset:0\n\t"
        "ds_load_b32 %1, %9 offset:4\n\t"
        "ds_load_b32 %2, %9 offset:8\n\t"
        "ds_load_b32 %3, %9 offset:12\n\t"
        "ds_load_b32 %4, %9 offset:16\n\t"
        "ds_load_b32 %5, %9 offset:20\n\t"
        "ds_load_b32 %6, %9 offset:24\n\t"
        "ds_load_b32 %7, %9 offset:28\n\t"
        "ds_load_b32 %8, %9 offset:32\n\t"
        "s_wait_dscnt 0"
        : "=&v"(m00), "=&v"(m01), "=&v"(m02),
          "=&v"(m10), "=&v"(m11), "=&v"(m12),
          "=&v"(m20), "=&v"(m21), "=&v"(m22)
        : "v"(lds_rd)
        : "memory");

    const size_t gid = (size_t)blk * TILE + tid;
    twist_math_store(m00, m01, m02, m10, m11, m12, m20, m21, m22, out, gid);
}

// Tail kernel: plain bounds-checked per-thread loads (only launched when
// nmat % TILE != 0; never for the B = 4,194,304 harness case).
__global__ void mat2twist_tail(const float* __restrict__ in, float* __restrict__ out,
                               int start, int nmat)
{
    int g = start + blockIdx.x * blockDim.x + threadIdx.x;
    if (g >= nmat) return;
    const float* m = in + (size_t)g * 9;
    twist_math_store(m[0], m[1], m[2], m[3], m[4], m[5], m[6], m[7], m[8],
                     out, (size_t)g);
}

extern "C" void kernel_launch(void* const* d_in, const int* in_sizes, int n_in,
                              void* d_out, int out_size, void* d_ws, size_t ws_size,
                              hipStream_t stream)
{
    (void)n_in; (void)out_size; (void)d_ws; (void)ws_size;
    const float* in  = (const float*)d_in[0];
    float*       out = (float*)d_out;

    const int nmat = in_sizes[0] / 9;        // 4,194,304
    const int full = nmat / TILE;            // 16,384 full tiles
    const int rem  = nmat - full * TILE;     // 0 for the harness shape

    if (full > 0)
        mat2twist_main<<<full, TPB, 0, stream>>>(in, out);
    if (rem > 0)
        mat2twist_tail<<<(rem + TPB - 1) / TPB, TPB, 0, stream>>>(in, out, full * TILE, nmat);
}